// GCNConv_47253230190757
// MI455X (gfx1250) — compile-verified
//
#include <hip/hip_runtime.h>

// ---------------------------------------------------------------------------
// GCN layer for MI455X (gfx1250, wave32):
//   z = A @ x   (COO SpMM, fp32 atomics, accumulated directly in d_out)
//   out = z @ W (in-place per 16-row tile, V_WMMA_F32_16X16X4_F32)
// Memory-bound problem: all operands fit in the 192MB L2, so the design
// optimizes L2-side access patterns (512B coalesced row gathers, LDS-staged
// A-fragments with bank-conflict-free padded stride) rather than FLOPs.
// ---------------------------------------------------------------------------

typedef float v2f __attribute__((ext_vector_type(2)));
typedef float v8f __attribute__((ext_vector_type(8)));

#define F_DIM 128
#define SZ_STRIDE 132   // padded LDS row stride: 132 mod 64 banks = 4 -> conflict-free strided reads

// --- SpMM: one wave32 per edge. 32 lanes x float4 = one full 128-float row. ---
__global__ __launch_bounds__(256) void spmm_edge_kernel(
    const int* __restrict__ row, const int* __restrict__ col,
    const float* __restrict__ vals, const float* __restrict__ x,
    float* __restrict__ z, int nEdges)
{
    const int e    = (int)((blockIdx.x * blockDim.x + threadIdx.x) >> 5);
    const int lane = threadIdx.x & 31;
    if (e >= nEdges) return;

    const int   r = row[e];
    const int   c = col[e];
    const float v = vals[e];

    // Coalesced 512B gather of source row (lanes cover 128 floats as float4).
    const float4 d = ((const float4*)(x + (size_t)c * F_DIM))[lane];

    float* dst = z + (size_t)r * F_DIM + lane * 4;
    atomicAdd(dst + 0, v * d.x);   // global_atomic_add_f32, resolved in L2
    atomicAdd(dst + 1, v * d.y);
    atomicAdd(dst + 2, v * d.z);
    atomicAdd(dst + 3, v * d.w);
}

// --- In-place projection: block = 16 rows x 128 cols, 8 waves, each wave one
//     16x16 tile via 32x V_WMMA_F32_16X16X4_F32 over K=128. ---
__global__ __launch_bounds__(256) void gemm_inplace_kernel(
    float* __restrict__ zout, const float* __restrict__ W, int nNodes)
{
    __shared__ float sZ[16 * SZ_STRIDE];   // 8.4 KB padded z tile

    const int  tid     = threadIdx.x;
    const long rowBase = (long)blockIdx.x * 16;

    // Stage the 16x128 z tile (coalesced reads; full tile read BEFORE any
    // store to the same rows -> in-place is safe).
    for (int i = tid; i < 16 * F_DIM; i += 256) {
        const int  r  = i >> 7;
        const int  cc = i & (F_DIM - 1);
        const long gr = rowBase + r;
        sZ[r * SZ_STRIDE + cc] = (gr < (long)nNodes) ? zout[gr * F_DIM + cc] : 0.0f;
    }
    __syncthreads();

    const int wave    = tid >> 5;
    const int lane    = tid & 31;
    const int m       = lane & 15;        // row within tile / col within B frag
    const int kh      = (lane >> 4) << 1; // 0 or 2: K-offset for upper half-wave
    const int colBase = wave << 4;        // 16-col slice of W per wave

    v8f acc = {};
#pragma unroll
    for (int k = 0; k < F_DIM; k += 4) {
        // A frag (16x4 f32): lane L -> row L&15; VGPR0/1 = K+kh, K+kh+1
        v2f a, b;
        a.x = sZ[m * SZ_STRIDE + k + kh];
        a.y = sZ[m * SZ_STRIDE + k + kh + 1];
        // B frag (4x16 f32): row K+kh / K+kh+1 of W, col = colBase+m.
        // W is 64KB -> resident in WGP$/L2, read directly per-lane.
        b.x = W[(k + kh)     * F_DIM + colBase + m];
        b.y = W[(k + kh + 1) * F_DIM + colBase + m];
        acc = __builtin_amdgcn_wmma_f32_16x16x4_f32(
            /*neg_a=*/false, a, /*neg_b=*/false, b,
            /*c_mod=*/(short)0, acc, /*reuse_a=*/false, /*reuse_b=*/false);
    }

    // D layout: VGPR i -> M = i + 8*(lane>=16), N = lane&15.
    const int mhi = (lane >> 4) << 3;
#pragma unroll
    for (int i = 0; i < 8; ++i) {
        const long gr = rowBase + mhi + i;
        if (gr < (long)nNodes) zout[gr * F_DIM + colBase + m] = acc[i];
    }
}

extern "C" void kernel_launch(void* const* d_in, const int* in_sizes, int n_in,
                              void* d_out, int out_size, void* d_ws, size_t ws_size,
                              hipStream_t stream)
{
    const int*   row  = (const int*)d_in[0];
    const int*   col  = (const int*)d_in[1];
    const float* vals = (const float*)d_in[2];
    const float* x    = (const float*)d_in[3];
    const float* W    = (const float*)d_in[4];
    float*       out  = (float*)d_out;

    const int nEdges = in_sizes[0];
    const int nNodes = in_sizes[3] / F_DIM;

    // Zero the accumulator (d_out doubles as z). Capture-safe memset node.
    hipMemsetAsync(out, 0, (size_t)out_size * sizeof(float), stream);

    // SpMM: 8 edges per 256-thread block (one wave32 per edge).
    const int gridS = (nEdges + 7) / 8;
    spmm_edge_kernel<<<gridS, 256, 0, stream>>>(row, col, vals, x, out, nEdges);

    // In-place WMMA projection: one block per 16 output rows.
    const int gridG = (nNodes + 15) / 16;
    gemm_inplace_kernel<<<gridG, 256, 0, stream>>>(out, W, nNodes);
}